// APPNP_Net_58583353918037
// MI455X (gfx1250) — compile-verified
//
#include <hip/hip_runtime.h>
#include <math.h>

#define NN      100000
#define EE      3200000
#define FIN     128
#define HIDF    64
#define NCLS    10
#define NG      512
#define KSTEPS  5
#define ALPHA_F 0.2f

typedef float v2f __attribute__((ext_vector_type(2)));
typedef float v8f __attribute__((ext_vector_type(8)));

__device__ __forceinline__ void atomicAddF(float* p, float v) {
    unsafeAtomicAdd(p, v);   // native global_atomic_add_f32
}

// ---------------------------------------------------------------------------
// h0 = relu(x @ W1 + b1)   via V_WMMA_F32_16X16X4_F32
// one wave per 16-row block; 4 column tiles of 16 (HID=64)
// ---------------------------------------------------------------------------
__global__ __launch_bounds__(32) void linear1_wmma(
        const float* __restrict__ x, const float* __restrict__ W1,
        const float* __restrict__ b1, float* __restrict__ h0) {
    const int row0  = blockIdx.x * 16;
    const int lane  = threadIdx.x;          // 0..31
    const int m     = lane & 15;
    const int khalf = (lane >> 4) << 1;     // 0 (lanes 0-15) or 2 (lanes 16-31)

    v8f acc0 = {}, acc1 = {}, acc2 = {}, acc3 = {};
    const float* xrow = x + (size_t)(row0 + m) * FIN;

    for (int k = 0; k < FIN; k += 4) {
        const int kk = k + khalf;
        v2f a; a.x = xrow[kk]; a.y = xrow[kk + 1];

        v2f b0; b0.x = W1[kk * HIDF + m +  0]; b0.y = W1[(kk + 1) * HIDF + m +  0];
        v2f b1v; b1v.x = W1[kk * HIDF + m + 16]; b1v.y = W1[(kk + 1) * HIDF + m + 16];
        v2f b2; b2.x = W1[kk * HIDF + m + 32]; b2.y = W1[(kk + 1) * HIDF + m + 32];
        v2f b3; b3.x = W1[kk * HIDF + m + 48]; b3.y = W1[(kk + 1) * HIDF + m + 48];

        acc0 = __builtin_amdgcn_wmma_f32_16x16x4_f32(false, a, false, b0,  (short)0, acc0, false, false);
        acc1 = __builtin_amdgcn_wmma_f32_16x16x4_f32(false, a, false, b1v, (short)0, acc1, false, false);
        acc2 = __builtin_amdgcn_wmma_f32_16x16x4_f32(false, a, false, b2,  (short)0, acc2, false, false);
        acc3 = __builtin_amdgcn_wmma_f32_16x16x4_f32(false, a, false, b3,  (short)0, acc3, false, false);
    }

    const int mbase = (lane < 16) ? 0 : 8;
    #pragma unroll
    for (int t = 0; t < 4; ++t) {
        const int   col  = t * 16 + m;
        const float bias = b1[col];
        const v8f   acc  = (t == 0) ? acc0 : (t == 1) ? acc1 : (t == 2) ? acc2 : acc3;
        #pragma unroll
        for (int j = 0; j < 8; ++j) {
            const int r = row0 + mbase + j;
            float v = acc[j] + bias;
            h0[(size_t)r * HIDF + col] = v > 0.f ? v : 0.f;
        }
    }
}

// ---------------------------------------------------------------------------
// degree / norm / per-edge coefficient
// ---------------------------------------------------------------------------
__global__ void deg_init(float* __restrict__ deg) {
    int i = blockIdx.x * blockDim.x + threadIdx.x;
    if (i < NN) deg[i] = 1.0f;                      // self-loop weight
}

__global__ void deg_accum(const int* __restrict__ ei, const float* __restrict__ ew,
                          float* __restrict__ deg) {
    int e = blockIdx.x * blockDim.x + threadIdx.x;
    if (e < EE) atomicAddF(&deg[ei[EE + e]], ew[e]);   // dst row
}

__global__ void dis_kernel(const float* __restrict__ deg, float* __restrict__ dis) {
    int i = blockIdx.x * blockDim.x + threadIdx.x;
    if (i < NN) { float d = deg[i]; dis[i] = d > 0.f ? rsqrtf(d) : 0.f; }
}

__global__ void coef_kernel(const int* __restrict__ ei, const float* __restrict__ ew,
                            const float* __restrict__ dis, float* __restrict__ coef) {
    int e = blockIdx.x * blockDim.x + threadIdx.x;
    if (e < EE) {
        int s = ei[e], d = ei[EE + e];
        coef[e] = (1.0f - ALPHA_F) * dis[s] * ew[e] * dis[d];
    }
}

// ---------------------------------------------------------------------------
// APPNP step: hn = (1-a)*(self-loop term) + a*x0, then scatter edges
// ---------------------------------------------------------------------------
__global__ void prop_init(const float* __restrict__ hcur, const float* __restrict__ h0,
                          const float* __restrict__ dis, float* __restrict__ hn) {
    int idx = blockIdx.x * blockDim.x + threadIdx.x;   // N*HID threads exactly
    int node = idx >> 6;
    float ds = dis[node];
    hn[idx] = (1.0f - ALPHA_F) * ds * ds * hcur[idx] + ALPHA_F * h0[idx];
}

__global__ void prop_scatter(const int* __restrict__ ei, const float* __restrict__ coef,
                             const float* __restrict__ hcur, float* __restrict__ hn) {
    int idx = blockIdx.x * blockDim.x + threadIdx.x;   // E*HID threads exactly
    int e = idx >> 6;
    int f = idx & 63;
    int s = ei[e], d = ei[EE + e];
    atomicAddF(&hn[d * HIDF + f], coef[e] * hcur[s * HIDF + f]);
}

// ---------------------------------------------------------------------------
// global mean pool
// ---------------------------------------------------------------------------
__global__ void pool_zero(float* __restrict__ sums, float* __restrict__ cnt) {
    int i = blockIdx.x * blockDim.x + threadIdx.x;     // NG*HID threads
    sums[i] = 0.f;
    if (i < NG) cnt[i] = 0.f;
}

__global__ void pool_accum(const float* __restrict__ h, const int* __restrict__ batch,
                           float* __restrict__ sums, float* __restrict__ cnt) {
    int idx = blockIdx.x * blockDim.x + threadIdx.x;   // N*HID threads
    int node = idx >> 6;
    int f = idx & 63;
    int g = batch[node];
    atomicAddF(&sums[g * HIDF + f], h[idx]);
    if (f == 0) atomicAddF(&cnt[g], 1.0f);
}

// ---------------------------------------------------------------------------
// head: pooled @ W2 + b2, log_softmax   (one 64-thread block per graph)
// ---------------------------------------------------------------------------
__global__ __launch_bounds__(64) void head_kernel(
        const float* __restrict__ sums, const float* __restrict__ cnt,
        const float* __restrict__ W2, const float* __restrict__ b2,
        float* __restrict__ out) {
    __shared__ float sp[HIDF];
    __shared__ float lg[NCLS];
    int g = blockIdx.x, t = threadIdx.x;
    sp[t] = sums[g * HIDF + t] / fmaxf(cnt[g], 1.0f);
    __syncthreads();
    if (t < NCLS) {
        float acc = b2[t];
        #pragma unroll 8
        for (int k = 0; k < HIDF; ++k) acc += sp[k] * W2[k * NCLS + t];
        lg[t] = acc;
    }
    __syncthreads();
    if (t < NCLS) {
        float mx = lg[0];
        #pragma unroll
        for (int c = 1; c < NCLS; ++c) mx = fmaxf(mx, lg[c]);
        float s = 0.f;
        #pragma unroll
        for (int c = 0; c < NCLS; ++c) s += expf(lg[c] - mx);
        out[g * NCLS + t] = lg[t] - mx - logf(s);
    }
}

// ---------------------------------------------------------------------------
extern "C" void kernel_launch(void* const* d_in, const int* in_sizes, int n_in,
                              void* d_out, int out_size, void* d_ws, size_t ws_size,
                              hipStream_t stream) {
    const float* x   = (const float*)d_in[0];
    const int*   ei  = (const int*)  d_in[1];   // [2, E]: row0 src, row1 dst
    const float* ew  = (const float*)d_in[2];
    const int*   bat = (const int*)  d_in[3];
    const float* W1  = (const float*)d_in[4];
    const float* b1  = (const float*)d_in[5];
    const float* W2  = (const float*)d_in[6];
    const float* b2  = (const float*)d_in[7];
    float* out = (float*)d_out;

    char* ws = (char*)d_ws;
    size_t off = 0;
    auto take = [&](size_t bytes) { char* p = ws + off; off = (off + bytes + 255) & ~(size_t)255; return p; };
    float* h0   = (float*)take((size_t)NN * HIDF * 4);
    float* hA   = (float*)take((size_t)NN * HIDF * 4);
    float* hB   = (float*)take((size_t)NN * HIDF * 4);
    float* deg  = (float*)take((size_t)NN * 4);
    float* dis  = (float*)take((size_t)NN * 4);
    float* coef = (float*)take((size_t)EE * 4);
    float* sums = (float*)take((size_t)NG * HIDF * 4);
    float* cnt  = (float*)take((size_t)NG * 4);

    // 1) h0 = relu(x @ W1 + b1)   [WMMA]
    linear1_wmma<<<NN / 16, 32, 0, stream>>>(x, W1, b1, h0);

    // 2) gcn_norm
    deg_init  <<<(NN + 255) / 256, 256, 0, stream>>>(deg);
    deg_accum <<<(EE + 255) / 256, 256, 0, stream>>>(ei, ew, deg);
    dis_kernel<<<(NN + 255) / 256, 256, 0, stream>>>(deg, dis);
    coef_kernel<<<(EE + 255) / 256, 256, 0, stream>>>(ei, ew, dis, coef);

    // 3) K APPNP propagation steps (double-buffered)
    float* hcur = h0;
    float* hnext = hA;
    for (int it = 0; it < KSTEPS; ++it) {
        prop_init   <<<(NN * HIDF) / 256, 256, 0, stream>>>(hcur, h0, dis, hnext);
        prop_scatter<<<(size_t)EE * HIDF / 256, 256, 0, stream>>>(ei, coef, hcur, hnext);
        if (it == 0) { hcur = hA; hnext = hB; }
        else         { float* t = hcur; hcur = hnext; hnext = t; }
    }

    // 4) global mean pool
    pool_zero <<<(NG * HIDF) / 256, 256, 0, stream>>>(sums, cnt);
    pool_accum<<<(NN * HIDF) / 256, 256, 0, stream>>>(hcur, bat, sums, cnt);

    // 5) head + log_softmax
    head_kernel<<<NG, 64, 0, stream>>>(sums, cnt, W2, b2, out);
}